// PatternDynamics_61486751809653
// MI455X (gfx1250) — compile-verified
//
#include <hip/hip_runtime.h>
#include <hip/hip_bf16.h>
#include <math.h>

typedef __attribute__((ext_vector_type(2))) float v2f;
typedef __attribute__((ext_vector_type(8))) float v8f;

// Problem dims
constexpr int  Bn = 4, Hn = 16, Sn = 1024, Dn = 256, Pn_ = 64;
constexpr size_t ROWS = (size_t)Bn * Hn * Sn;            // 65536 rows of states
// Flat output layout (tuple concatenated in return order)
constexpr size_t OFF_ROUTING = 0;
constexpr size_t N_ROUTING   = ROWS;                     // (4,16,1024)
constexpr size_t OFF_ROUTES  = OFF_ROUTING + N_ROUTING;
constexpr size_t N_ROUTES    = ROWS * (size_t)Sn;        // (4,16,1024,1024)
constexpr size_t OFF_TW      = OFF_ROUTES + N_ROUTES;
constexpr size_t N_TW        = (size_t)16 * 64 * 64;     // (16,64,64)
constexpr size_t OFF_PS      = OFF_TW + N_TW;
constexpr size_t N_PS        = ROWS * (size_t)Pn_;       // (4,16,1024,64)

// ---------------------------------------------------------------------------
// 1) routing_scores = softmax(const) = 1/1024 everywhere  (float4 fill)
__global__ void k_fill_routing(float* __restrict__ out) {
    size_t i = (size_t)blockIdx.x * blockDim.x + threadIdx.x;   // float4 index
    float4 v = make_float4(1.0f/1024.0f, 1.0f/1024.0f, 1.0f/1024.0f, 1.0f/1024.0f);
    ((float4*)out)[i] = v;
}

// ---------------------------------------------------------------------------
// 2) optimal_routes[b,h,i,j] = -j/1024  (268 MB bandwidth fill, float4 stores)
__global__ void k_fill_routes(float* __restrict__ out) {
    size_t i = (size_t)blockIdx.x * blockDim.x + threadIdx.x;   // float4 index
    int j = (int)((i * 4) & (size_t)(Sn - 1));                  // row offset (mult of 4)
    const float sc = -1.0f / 1024.0f;
    float4 v = make_float4(j * sc, (j + 1) * sc, (j + 2) * sc, (j + 3) * sc);
    ((float4*)out)[i] = v;
}

// ---------------------------------------------------------------------------
// 3) transfer_weights passthrough
__global__ void k_copy_tw(const float* __restrict__ src, float* __restrict__ dst) {
    size_t i = (size_t)blockIdx.x * blockDim.x + threadIdx.x;   // float4 index
    ((float4*)dst)[i] = ((const float4*)src)[i];
}

// ---------------------------------------------------------------------------
// 4) normalize patterns, fold in 1/TEMPERATURE: pn = patterns * (10 / max(||row||, eps))
__global__ void k_norm_patterns(const float* __restrict__ patterns, float* __restrict__ pn) {
    __shared__ float red[256];
    const int row = blockIdx.x;          // 64 rows
    const int tid = threadIdx.x;         // 256 threads = d
    float v = patterns[(size_t)row * Dn + tid];
    red[tid] = v * v;
    __syncthreads();
    #pragma unroll
    for (int s = 128; s > 0; s >>= 1) {
        if (tid < s) red[tid] += red[tid + s];
        __syncthreads();
    }
    float scale = 10.0f / fmaxf(sqrtf(red[0]), 1e-12f);
    pn[(size_t)row * Dn + tid] = v * scale;
}

// ---------------------------------------------------------------------------
// 5) logits GEMM via V_WMMA_F32_16X16X4_F32.
// One wave computes a 16x64 strip: 4 16x16 accumulator tiles, A fragment reused.
// A 16x4 f32 layout: lane L holds A[L&15][2*(L>>4)+{0,1}]  (float2, contiguous)
// B  4x16 f32 layout: lane L holds B[2*(L>>4)+{0,1}][L&15] = PnT -> pn row (L&15), same floats
// D 16x16 f32 layout: vgpr v, lane L -> D[v + 8*(L>>4)][L&15]
__global__ void k_gemm_wmma(const float* __restrict__ states,
                            const float* __restrict__ pn,
                            float* __restrict__ ps) {
    const int wave    = threadIdx.x >> 5;
    const int lane    = threadIdx.x & 31;
    const int tile    = blockIdx.x * 8 + wave;        // 4096 row tiles total
    const int rowBase = tile * 16;
    const int m  = lane & 15;
    const int kh = lane >> 4;

    const float* arow = states + (size_t)(rowBase + m) * Dn + 2 * kh;
    const float* b0   = pn + (size_t)( 0 + m) * Dn + 2 * kh;
    const float* b1   = pn + (size_t)(16 + m) * Dn + 2 * kh;
    const float* b2   = pn + (size_t)(32 + m) * Dn + 2 * kh;
    const float* b3   = pn + (size_t)(48 + m) * Dn + 2 * kh;

    v8f acc0 = {}, acc1 = {}, acc2 = {}, acc3 = {};

    #pragma unroll 4
    for (int k0 = 0; k0 < Dn; k0 += 4) {
        v2f a   = *(const v2f*)(arow + k0);
        v2f vb0 = *(const v2f*)(b0 + k0);
        v2f vb1 = *(const v2f*)(b1 + k0);
        v2f vb2 = *(const v2f*)(b2 + k0);
        v2f vb3 = *(const v2f*)(b3 + k0);
        acc0 = __builtin_amdgcn_wmma_f32_16x16x4_f32(false, a, false, vb0, (short)0, acc0, false, false);
        acc1 = __builtin_amdgcn_wmma_f32_16x16x4_f32(false, a, false, vb1, (short)0, acc1, false, false);
        acc2 = __builtin_amdgcn_wmma_f32_16x16x4_f32(false, a, false, vb2, (short)0, acc2, false, false);
        acc3 = __builtin_amdgcn_wmma_f32_16x16x4_f32(false, a, false, vb3, (short)0, acc3, false, false);
    }

    float* out = ps + (size_t)rowBase * Pn_;
    #pragma unroll
    for (int v = 0; v < 8; ++v) {
        size_t r = (size_t)(v + 8 * kh) * Pn_ + m;
        out[r +  0] = acc0[v];
        out[r + 16] = acc1[v];
        out[r + 32] = acc2[v];
        out[r + 48] = acc3[v];
    }
}

// ---------------------------------------------------------------------------
// 6) in-place row softmax over 64 logits (one thread per row)
__global__ void k_softmax64(float* __restrict__ ps) {
    size_t r = (size_t)blockIdx.x * blockDim.x + threadIdx.x;   // 65536 rows
    float4* row = (float4*)(ps + r * (size_t)Pn_);
    float4 v[16];
    float mx = -3.4e38f;
    #pragma unroll
    for (int i = 0; i < 16; ++i) {
        v[i] = row[i];
        mx = fmaxf(mx, fmaxf(fmaxf(v[i].x, v[i].y), fmaxf(v[i].z, v[i].w)));
    }
    float sum = 0.0f;
    #pragma unroll
    for (int i = 0; i < 16; ++i) {
        v[i].x = __expf(v[i].x - mx);
        v[i].y = __expf(v[i].y - mx);
        v[i].z = __expf(v[i].z - mx);
        v[i].w = __expf(v[i].w - mx);
        sum += v[i].x + v[i].y + v[i].z + v[i].w;
    }
    float inv = 1.0f / sum;
    #pragma unroll
    for (int i = 0; i < 16; ++i) {
        v[i].x *= inv; v[i].y *= inv; v[i].z *= inv; v[i].w *= inv;
        row[i] = v[i];
    }
}

// ---------------------------------------------------------------------------
extern "C" void kernel_launch(void* const* d_in, const int* in_sizes, int n_in,
                              void* d_out, int out_size, void* d_ws, size_t ws_size,
                              hipStream_t stream) {
    const float* states   = (const float*)d_in[0];
    const float* patterns = (const float*)d_in[1];
    const float* tw       = (const float*)d_in[2];
    float* out = (float*)d_out;
    float* pn  = (float*)d_ws;                 // 64*256 f32 = 64 KB scratch

    float* routing = out + OFF_ROUTING;
    float* routes  = out + OFF_ROUTES;
    float* twout   = out + OFF_TW;
    float* ps      = out + OFF_PS;

    // fills / copy (pure bandwidth)
    k_fill_routing<<<(unsigned)(N_ROUTING / 4 / 256), 256, 0, stream>>>(routing);
    k_fill_routes <<<(unsigned)(N_ROUTES  / 4 / 256), 256, 0, stream>>>(routes);
    k_copy_tw     <<<(unsigned)(N_TW      / 4 / 256), 256, 0, stream>>>(tw, twout);

    // pattern normalization (folds 1/TEMPERATURE)
    k_norm_patterns<<<Pn_, Dn, 0, stream>>>(patterns, pn);

    // WMMA GEMM: 4096 row tiles / 8 waves per block = 512 blocks
    k_gemm_wmma<<<512, 256, 0, stream>>>(states, pn, ps);

    // softmax over p=64, in place
    k_softmax64<<<(unsigned)(ROWS / 256), 256, 0, stream>>>(ps);
}